// TemporalGNN_43482248904963
// MI455X (gfx1250) — compile-verified
//
#include <hip/hip_runtime.h>
#include <math.h>

typedef __attribute__((ext_vector_type(2))) float v2f;
typedef __attribute__((ext_vector_type(8))) float v8f;

#define N_NODES 20000
#define F_INV   8
#define T_P     12
#define N_EDGES 320000
#define HIDV    32

// workspace layout (float offsets)
#define OFF_MCAT  0u                         // 8 x 64 fused (Wz@LzTop | Wh@LhTop), Mcat[k*64+j]
#define OFF_CB    512u                       // 64 fused biases (cz | ch)
#define OFF_PROBS 576u                       // 12 softmax probs (pad 16)
#define OFF_DEG   640u                       // N degrees
#define OFF_DINV  (OFF_DEG + N_NODES)        // N rsqrt(deg+1)
#define OFF_XAGG  (OFF_DINV + N_NODES)       // N x 96 aggregated features, layout (f,t): n*96 + f*12 + t
#define OFF_HACC  (OFF_XAGG + N_NODES * 96u) // N x 32 attention-pooled hidden

__device__ __forceinline__ float fast_rcp(float x) {
#if __has_builtin(__builtin_amdgcn_rcpf)
    return __builtin_amdgcn_rcpf(x);
#else
    return 1.0f / x;
#endif
}

// returns 1 - sigmoid(a) = 1/(1+exp(a)); branch-free
__device__ __forceinline__ float one_minus_sigmoid(float a) {
    return fast_rcp(1.0f + __expf(a));
}

// branch-free tanh; uses CDNA5 hardware V_TANH_F32 when available
__device__ __forceinline__ float fast_tanh(float x) {
#if __has_builtin(__builtin_amdgcn_tanhf)
    return __builtin_amdgcn_tanhf(x);
#else
    float e = __expf(-2.0f * x);
    return (1.0f - e) * fast_rcp(1.0f + e);
#endif
}

__global__ void k_zero(float* ws, unsigned base, unsigned count) {
    unsigned i = blockIdx.x * blockDim.x + threadIdx.x;
    if (i < count) ws[base + i] = 0.0f;
}

// Fold Wz@Lz[:32], Wh@Lh[:32], biases, and softmax(att) into ws.
__global__ void k_prep(const float* __restrict__ Wz, const float* __restrict__ bz,
                       const float* __restrict__ Wh, const float* __restrict__ bh,
                       const float* __restrict__ Lz, const float* __restrict__ lz,
                       const float* __restrict__ Lh, const float* __restrict__ lh,
                       const float* __restrict__ att, float* __restrict__ ws) {
    int tid = threadIdx.x;
    if (tid < 512) {                           // Mcat: 8 x 64
        int k = tid >> 6, j = tid & 63;
        const float* W = (j < 32) ? Wz : Wh;
        const float* L = (j < 32) ? Lz : Lh;
        int h = j & 31;
        float s = 0.0f;
        for (int c = 0; c < 32; ++c) s += W[k * 32 + c] * L[c * 32 + h];
        ws[OFF_MCAT + k * 64 + j] = s;
    } else if (tid < 576) {                    // fused biases: b@L_top + l
        int j = tid - 512;
        const float* b = (j < 32) ? bz : bh;
        const float* L = (j < 32) ? Lz : Lh;
        const float* l = (j < 32) ? lz : lh;
        int h = j & 31;
        float s = l[h];
        for (int c = 0; c < 32; ++c) s += b[c] * L[c * 32 + h];
        ws[OFF_CB + j] = s;
    } else if (tid == 576) {                   // softmax over periods
        float m = att[0];
        for (int t = 1; t < T_P; ++t) m = fmaxf(m, att[t]);
        float e[T_P], den = 0.0f;
        for (int t = 0; t < T_P; ++t) { e[t] = __expf(att[t] - m); den += e[t]; }
        for (int t = 0; t < T_P; ++t) ws[OFF_PROBS + t] = e[t] / den;
    }
}

__global__ void k_deg(const int* __restrict__ dst, float* __restrict__ ws) {
    int e = blockIdx.x * blockDim.x + threadIdx.x;
    if (e < N_EDGES) atomicAdd(&ws[OFF_DEG + dst[e]], 1.0f);
}

__global__ void k_dinv(float* __restrict__ ws) {
    int n = blockIdx.x * blockDim.x + threadIdx.x;
    if (n < N_NODES) ws[OFF_DINV + n] = rsqrtf(ws[OFF_DEG + n] + 1.0f);
}

// Scatter raw features along edges: xagg[dst] += dinv[src]*dinv[dst] * x[src]
// 2 edges per 192-thread block; k = f*12 + t matches x's native layout -> coalesced.
__global__ void k_edge(const int* __restrict__ ei, const float* __restrict__ x,
                       float* __restrict__ ws) {
    int local = threadIdx.x;
    int e = blockIdx.x * 2 + (local >= 96 ? 1 : 0);
    int k = local - 96 * (local >= 96 ? 1 : 0);
    if (e >= N_EDGES) return;
    int s = ei[e];
    int d = ei[N_EDGES + e];
    float norm = ws[OFF_DINV + s] * ws[OFF_DINV + d];
    float v = norm * x[(size_t)s * 96 + k];
    atomicAdd(&ws[OFF_XAGG + (size_t)d * 96 + k], v);
}

// Self-loop: xagg[n] += dinv[n]^2 * x[n]
__global__ void k_self(const float* __restrict__ x, float* __restrict__ ws) {
    unsigned g = blockIdx.x * blockDim.x + threadIdx.x;
    if (g >= N_NODES * 96u) return;
    unsigned n = g / 96u;
    float di = ws[OFF_DINV + n];
    ws[OFF_XAGG + g] += di * di * x[g];
}

// Per wave: 16 nodes. For each period t, compute y = xagg_tile(16x8) @ Mcat(8x64)
// with 8 f32 WMMAs (4 col tiles x 2 K-halves), apply gates branch-free, and
// accumulate attention-weighted Hnew into registers. Store H_accum (no atomics).
__global__ void __launch_bounds__(128) k_gate(float* __restrict__ ws) {
    const float* Mcat  = ws + OFF_MCAT;
    const float* cb    = ws + OFF_CB;
    const float* probs = ws + OFF_PROBS;
    const float* xagg  = ws + OFF_XAGG;
    float* hacc        = ws + OFF_HACC;

    int lane = threadIdx.x & 31;
    int wave = threadIdx.x >> 5;
    int half = lane >> 4;       // lane group (K split for A/B, M split for D)
    int lm   = lane & 15;
    int n0   = (blockIdx.x * 4 + wave) * 16;

    // B tiles: B[j][s] covers output cols 16j..16j+15, K = 4s..4s+3.
    // Layout: VGPR0 = K row {2*half}, VGPR1 = K row {2*half+1}, lane lm = column.
    v2f B[4][2];
    for (int j = 0; j < 4; ++j)
        for (int s2 = 0; s2 < 2; ++s2) {
            int kb = 4 * s2 + 2 * half;
            int col = 16 * j + lm;
            B[j][s2].x = Mcat[(kb + 0) * 64 + col];
            B[j][s2].y = Mcat[(kb + 1) * 64 + col];
        }
    float cbz0 = cb[lm], cbz1 = cb[16 + lm];
    float cbh0 = cb[32 + lm], cbh1 = cb[48 + lm];

    int arow = n0 + lm; if (arow > N_NODES - 1) arow = N_NODES - 1; // clamp, EXEC stays full
    const float* ap = xagg + (size_t)arow * 96;

    v8f acc0 = {}; v8f acc1 = {};
    for (int t = 0; t < T_P; ++t) {
        // A tiles: lane lm = row M; K = (4*s + 2*half + {0,1}); element (f=K) at f*12+t
        v2f a0, a1;
        a0.x = ap[(2 * half + 0) * 12 + t];
        a0.y = ap[(2 * half + 1) * 12 + t];
        a1.x = ap[(4 + 2 * half + 0) * 12 + t];
        a1.y = ap[(4 + 2 * half + 1) * 12 + t];

        v8f cz0 = {}, cz1 = {}, ch0 = {}, ch1 = {};
        cz0 = __builtin_amdgcn_wmma_f32_16x16x4_f32(false, a0, false, B[0][0], (short)0, cz0, false, false);
        cz0 = __builtin_amdgcn_wmma_f32_16x16x4_f32(false, a1, false, B[0][1], (short)0, cz0, false, false);
        cz1 = __builtin_amdgcn_wmma_f32_16x16x4_f32(false, a0, false, B[1][0], (short)0, cz1, false, false);
        cz1 = __builtin_amdgcn_wmma_f32_16x16x4_f32(false, a1, false, B[1][1], (short)0, cz1, false, false);
        ch0 = __builtin_amdgcn_wmma_f32_16x16x4_f32(false, a0, false, B[2][0], (short)0, ch0, false, false);
        ch0 = __builtin_amdgcn_wmma_f32_16x16x4_f32(false, a1, false, B[2][1], (short)0, ch0, false, false);
        ch1 = __builtin_amdgcn_wmma_f32_16x16x4_f32(false, a0, false, B[3][0], (short)0, ch1, false, false);
        ch1 = __builtin_amdgcn_wmma_f32_16x16x4_f32(false, a1, false, B[3][1], (short)0, ch1, false, false);

        float p = probs[t];
        for (int i = 0; i < 8; ++i) {
            // Hnew = (1 - sigmoid(gz)) * tanh(gh); both branch-free (keeps EXEC full)
            float omz0 = one_minus_sigmoid(cz0[i] + cbz0);
            float h0   = fast_tanh(ch0[i] + cbh0);
            acc0[i] += p * omz0 * h0;
            float omz1 = one_minus_sigmoid(cz1[i] + cbz1);
            float h1   = fast_tanh(ch1[i] + cbh1);
            acc1[i] += p * omz1 * h1;
        }
    }
    // D layout: lane lm = col N, row M = i + 8*half
    for (int i = 0; i < 8; ++i) {
        int row = n0 + i + 8 * half;
        if (row < N_NODES) {
            hacc[row * 32 + lm]      = acc0[i];
            hacc[row * 32 + 16 + lm] = acc1[i];
        }
    }
}

// out = relu(H_accum) @ Wout + bout via 8 f32 WMMAs per wave (K=32, N padded 12->16)
__global__ void __launch_bounds__(128) k_out(const float* __restrict__ Wout,
                                             const float* __restrict__ bout,
                                             const float* __restrict__ ws,
                                             float* __restrict__ out) {
    const float* hacc = ws + OFF_HACC;
    int lane = threadIdx.x & 31;
    int wave = threadIdx.x >> 5;
    int half = lane >> 4;
    int lm   = lane & 15;
    int n0   = (blockIdx.x * 4 + wave) * 16;

    int arow = n0 + lm; if (arow > N_NODES - 1) arow = N_NODES - 1;
    const float* ap = hacc + (size_t)arow * 32;

    v8f c = {};
    for (int kb = 0; kb < 32; kb += 4) {
        v2f a, b;
        a.x = fmaxf(ap[kb + 2 * half + 0], 0.0f);
        a.y = fmaxf(ap[kb + 2 * half + 1], 0.0f);
        b.x = (lm < T_P) ? Wout[(kb + 2 * half + 0) * T_P + lm] : 0.0f;
        b.y = (lm < T_P) ? Wout[(kb + 2 * half + 1) * T_P + lm] : 0.0f;
        c = __builtin_amdgcn_wmma_f32_16x16x4_f32(false, a, false, b, (short)0, c, false, false);
    }
    float bo = (lm < T_P) ? bout[lm] : 0.0f;
    for (int i = 0; i < 8; ++i) {
        int row = n0 + i + 8 * half;
        if (row < N_NODES && lm < T_P) out[row * T_P + lm] = c[i] + bo;
    }
}

extern "C" void kernel_launch(void* const* d_in, const int* in_sizes, int n_in,
                              void* d_out, int out_size, void* d_ws, size_t ws_size,
                              hipStream_t stream) {
    const float* x    = (const float*)d_in[0];
    const int*   ei   = (const int*)d_in[1];
    const float* Wz   = (const float*)d_in[2];
    const float* bz   = (const float*)d_in[3];
    // d_in[4] (Wr), d_in[5] (br), d_in[10] (Lr), d_in[11] (lr): dead code (H0==0)
    const float* Wh   = (const float*)d_in[6];
    const float* bh   = (const float*)d_in[7];
    const float* Lz   = (const float*)d_in[8];
    const float* lz   = (const float*)d_in[9];
    const float* Lh   = (const float*)d_in[12];
    const float* lh   = (const float*)d_in[13];
    const float* att  = (const float*)d_in[14];
    const float* Wout = (const float*)d_in[15];
    const float* bout = (const float*)d_in[16];
    float* ws  = (float*)d_ws;
    float* out = (float*)d_out;

    unsigned zc = (OFF_XAGG + N_NODES * 96u) - OFF_DEG; // deg + dinv + xagg
    k_zero<<<(zc + 255) / 256, 256, 0, stream>>>(ws, OFF_DEG, zc);
    k_prep<<<1, 640, 0, stream>>>(Wz, bz, Wh, bh, Lz, lz, Lh, lh, att, ws);
    k_deg<<<(N_EDGES + 255) / 256, 256, 0, stream>>>(ei + N_EDGES, ws);
    k_dinv<<<(N_NODES + 255) / 256, 256, 0, stream>>>(ws);
    k_edge<<<(N_EDGES + 1) / 2, 192, 0, stream>>>(ei, x, ws);
    k_self<<<(N_NODES * 96u + 255) / 256, 256, 0, stream>>>(x, ws);
    k_gate<<<(N_NODES + 63) / 64, 128, 0, stream>>>(ws);
    k_out<<<(N_NODES + 63) / 64, 128, 0, stream>>>(Wout, bout, ws, out);
}